// hausdorff_loss_with_intensity_63127429316932
// MI455X (gfx1250) — compile-verified
//
#include <hip/hip_runtime.h>
#include <hip/hip_bf16.h>

typedef float v2f __attribute__((ext_vector_type(2)));
typedef float v4f __attribute__((ext_vector_type(4)));
typedef float v8f __attribute__((ext_vector_type(8)));

#define INTENSITY_WEIGHT 0.5f
#define LOSS_WEIGHT 1.0f
#define NCHUNK 8      // column chunks; grid = (N/64) * NCHUNK = 1024 waves
#define RTILES 4      // row tiles (of 16 rows) per wave

// Raw min3 keeps LLVM from inserting canonicalize (v_max x,x) ops on the
// WMMA outputs; scheduler still sees the register dependencies.
#define MIN3_ACC(acc, x, y) \
  asm("v_min3_num_f32 %0, %0, %1, %2" : "+v"(acc) : "v"(x), "v"(y))

// ---- Kernel 0: per-column records {q0,q1,b2,0} {q2,0.5*q3,b2,0} -------------
__global__ void prep_kernel(const float* __restrict__ B, float* __restrict__ rec,
                            int M)
{
  const int m = blockIdx.x * blockDim.x + threadIdx.x;
  if (m >= M) return;
  const v4f q = *(const v4f*)(B + (size_t)m * 4);
  const float wq = INTENSITY_WEIGHT * q.w;
  const float b2 = q.x * q.x + q.y * q.y + q.z * q.z + wq * wq;
  v4f r0 = {q.x, q.y, b2, 0.0f};
  v4f r1 = {q.z, wq,  b2, 0.0f};
  *(v4f*)(rec + (size_t)m * 8)     = r0;
  *(v4f*)(rec + (size_t)m * 8 + 4) = r1;
}

// ---- Kernel 1: one wave32 = 4 row tiles (64 rows) x one column chunk. -------
// A fragment holds -2*a (intensity-scaled); b2 rides in through the WMMA C
// operand, so each WMMA emits d = b2 - 2 a.b directly. Loop body per 2 column
// tiles: 2 b96 loads, 2 C broadcasts, 8 WMMAs, 32 inline min3.
__global__ __launch_bounds__(32) void hausdorff_wmma_kernel(
    const float* __restrict__ A,     // [N,4] adv_pc
    const float* __restrict__ rec,   // [M,8] column records
    float* __restrict__ part,        // [NCHUNK, N] partial per-row mins
    int N, int M)
{
  const int lane = threadIdx.x;        // 0..31
  const int half = lane >> 4;          // 0: K=0,1 ; 1: K=2,3
  const int l16  = lane & 15;
  const int rg   = blockIdx.x;         // row group of 64 rows (4 tiles)
  const int cc   = blockIdx.y;         // column chunk
  const int mchunk = M / NCHUNK;
  const int m0 = cc * mchunk;

  // A fragments (16x4 f32): lane L<16 -> K=0,1 ; L>=16 -> K=2,3 (scaled),
  // with the -2 of the sqdist expansion folded in.
  v2f a[RTILES];
#pragma unroll
  for (int t = 0; t < RTILES; ++t) {
    const float* p = A + (size_t)(rg * 64 + t * 16 + l16) * 4 + half * 2;
    float ax = p[0], ay = p[1];
    if (half) ay *= INTENSITY_WEIGHT;
    a[t].x = -2.0f * ax;
    a[t].y = -2.0f * ay;
  }

  float acc[RTILES][8];
#pragma unroll
  for (int t = 0; t < RTILES; ++t)
#pragma unroll
    for (int v = 0; v < 8; ++v) acc[t][v] = 3.4e38f;

  const float* p = rec + (size_t)(m0 + l16) * 8 + half * 4;
  for (int it = 0; it < mchunk / 32; ++it, p += 32 * 8) { // 2 column tiles/iter
    const v4f qa = *(const v4f*)p;              // {bK0, bK1, b2, -}
    const v4f qb = *(const v4f*)(p + 16 * 8);
    const v2f bA = {qa.x, qa.y};
    const v2f bB = {qb.x, qb.y};

    // broadcast b2 into the C operand (shared by all 4 row tiles)
    v8f ciA, ciB;
#pragma unroll
    for (int v = 0; v < 8; ++v) { ciA[v] = qa.z; ciB[v] = qb.z; }

    v8f cA[RTILES], cB[RTILES];
#pragma unroll
    for (int t = 0; t < RTILES; ++t) {
      cA[t] = __builtin_amdgcn_wmma_f32_16x16x4_f32(false, a[t], false, bA,
                                                    (short)0, ciA, false, false);
      cB[t] = __builtin_amdgcn_wmma_f32_16x16x4_f32(false, a[t], false, bB,
                                                    (short)0, ciB, false, false);
    }

    // acc = min3(acc, dA, dB) with d = b2 - 2 a.b straight out of the WMMA
#pragma unroll
    for (int t = 0; t < RTILES; ++t)
#pragma unroll
      for (int v = 0; v < 8; ++v) {
        float dA = cA[t][v], dB = cB[t][v];
        MIN3_ACC(acc[t][v], dA, dB);
      }
  }

  // min across the 16 lanes of each half (xor<16 stays inside a half)
#pragma unroll
  for (int t = 0; t < RTILES; ++t)
#pragma unroll
    for (int v = 0; v < 8; ++v) {
      float x = acc[t][v];
      x = fminf(x, __shfl_xor(x, 1, 32));
      x = fminf(x, __shfl_xor(x, 2, 32));
      x = fminf(x, __shfl_xor(x, 4, 32));
      x = fminf(x, __shfl_xor(x, 8, 32));
      acc[t][v] = x;      // rowmin of row (t*16 + half*8 + v)
    }

  // lanes 0 and 16 each own 8 consecutive rows per tile -> contiguous stores
  if (l16 == 0) {
    float* w = part + (size_t)cc * N + rg * 64 + half * 8;
#pragma unroll
    for (int t = 0; t < RTILES; ++t)
#pragma unroll
      for (int v = 0; v < 8; ++v) w[t * 16 + v] = acc[t][v];
  }
}

// ---- Kernel 2: min over chunks, add ||a||^2 per row, global max. ------------
__global__ void reduce_max_kernel(const float* __restrict__ A,
                                  const float* __restrict__ part,
                                  float* __restrict__ out, int N)
{
  __shared__ float smem[256];
  const int t = threadIdx.x;
  float mx = -3.4e38f;
  for (int r = t; r < N; r += 256) {
    float m = 3.4e38f;
#pragma unroll
    for (int c = 0; c < NCHUNK; ++c) m = fminf(m, part[(size_t)c * N + r]);
    const v4f p = *(const v4f*)(A + (size_t)r * 4);
    const float w = INTENSITY_WEIGHT * p.w;
    const float a2 = p.x * p.x + p.y * p.y + p.z * p.z + w * w;
    mx = fmaxf(mx, a2 + m);
  }
  smem[t] = mx;
  __syncthreads();
  for (int s = 128; s > 0; s >>= 1) {
    if (t < s) smem[t] = fmaxf(smem[t], smem[t + s]);
    __syncthreads();
  }
  if (t == 0) out[0] = smem[0] * LOSS_WEIGHT;
}

extern "C" void kernel_launch(void* const* d_in, const int* in_sizes, int n_in,
                              void* d_out, int out_size, void* d_ws, size_t ws_size,
                              hipStream_t stream) {
  const float* adv = (const float*)d_in[0];  // [N,4]
  const float* ori = (const float*)d_in[1];  // [M,4]
  float* out = (float*)d_out;                // [1]

  const int N = in_sizes[0] / 4;
  const int M = in_sizes[1] / 4;

  float* rec  = (float*)d_ws;                // 8*M floats     (256 KB)
  float* part = rec + (size_t)8 * M;         // NCHUNK*N floats (256 KB)

  prep_kernel<<<(M + 255) / 256, 256, 0, stream>>>(ori, rec, M);
  dim3 grid(N / (16 * RTILES), NCHUNK);
  hausdorff_wmma_kernel<<<grid, 32, 0, stream>>>(adv, rec, part, N, M);
  reduce_max_kernel<<<1, 256, 0, stream>>>(adv, part, out, N);
}